// DiffPhysKAN_81415400063381
// MI455X (gfx1250) — compile-verified
//
#include <hip/hip_runtime.h>
#include <hip/hip_bf16.h>
#include <math.h>

typedef __attribute__((ext_vector_type(2))) float v2f;
typedef __attribute__((ext_vector_type(8))) float v8f;

#define CHUNK    1024
#define HALO_MAX 128
#define EXT_MAX  (CHUNK + 2 * HALO_MAX)

// ---------------------------------------------------------------------------
// Kernel 1: KAN MLP -> betas[T].  Layer-1 matmul (basis[16x20] @ w1[20x16])
// done with V_WMMA_F32_16X16X4_F32, 5 K-chunks of 4 accumulated in v8f.
// One wave per 16-row M-tile; layer 2 + softplus scalar per row.
// ---------------------------------------------------------------------------
__global__ __launch_bounds__(256) void kan_beta_kernel(
    const float* __restrict__ t_steps, const float* __restrict__ grid1,
    const float* __restrict__ sw1,     const float* __restrict__ bw1,
    const float* __restrict__ grid2,   const float* __restrict__ sw2,
    const float* __restrict__ bw2,     float* __restrict__ betas,
    int T, int G)
{
    __shared__ float hbuf[HALO_MAX * 16];   // h[T,16], T <= 128

    const int tid  = threadIdx.x;
    const int wave = tid >> 5;
    const int lane = tid & 31;
    const int half = lane >> 4;   // 0: K pair {k0,k0+1}, 1: {k0+2,k0+3}
    const int col  = lane & 15;   // M for A, N for B/C/D
    const int r0   = wave * 16;

    if (r0 < T) {
        // A-matrix 16x4 f32 layout: lane L holds M = L&15,
        // VGPR0 = K = k0 + 2*half, VGPR1 = K = k0 + 2*half + 1.
        const int arow  = r0 + col;
        const float tb  = t_steps[arow < T ? arow : (T - 1)];

        v8f c = {};
        for (int k0 = 0; k0 < G; k0 += 4) {
            const int ka = k0 + half * 2;
            v2f a, b;
            const float d0 = tb - grid1[ka];
            const float d1 = tb - grid1[ka + 1];
            a.x = expf(-10.0f * d0 * d0);
            a.y = expf(-10.0f * d1 * d1);
            // B-matrix 4x16 f32: same K-split across lane halves, N = col.
            b.x = sw1[ka * 16 + col];
            b.y = sw1[(ka + 1) * 16 + col];
            c = __builtin_amdgcn_wmma_f32_16x16x4_f32(
                    false, a, false, b, (short)0, c, false, false);
        }
        // C/D layout: VGPR r, lanes 0-15 -> row r0+r, lanes 16-31 -> r0+r+8.
        for (int r = 0; r < 8; ++r) {
            const int row = r0 + r + half * 8;
            if (row < T) {
                hbuf[row * 16 + col] = c[r] + t_steps[row] * bw1[col];
            }
        }
    }
    __syncthreads();

    if (tid < T) {
        float hrow[16];
        float acc = 0.0f;
        for (int i = 0; i < 16; ++i) {
            hrow[i] = hbuf[tid * 16 + i];
            acc += hrow[i] * bw2[i];                 // base: h @ base_w2.T
        }
        for (int i = 0; i < 16; ++i) {
            const float h = hrow[i];
            for (int g = 0; g < G; ++g) {
                const float d = h - grid2[g];
                acc += expf(-10.0f * d * d) * sw2[i * G + g];
            }
        }
        const float sp = (acc > 20.0f) ? acc : log1pf(expf(acc)); // softplus
        betas[tid] = fminf(fmaxf(sp, 0.0f), 20.0f);
    }
}

// ---------------------------------------------------------------------------
// Kernel 2: temporally-blocked PDE integrator.  Each block owns CHUNK cells,
// loads CHUNK + 2T cells into LDS once, runs all T steps locally with a
// double-buffered LDS ping-pong (1 barrier/step), streams each step's owned
// cells to history with non-temporal stores.  Correct region shrinks 1
// cell/step from each buffer edge; halo = T keeps the owned chunk exact.
// ---------------------------------------------------------------------------
__global__ __launch_bounds__(256) void pde_history_kernel(
    const float* __restrict__ I0,     const float* __restrict__ betas,
    const float* __restrict__ t_steps, const float* __restrict__ x_grid,
    const float* __restrict__ diff_param, float* __restrict__ history,
    int N, int T)
{
    __shared__ float buf[2][EXT_MAX];
    __shared__ float sbeta[HALO_MAX];

    const int tid  = threadIdx.x;
    const int halo = T;                       // T <= HALO_MAX by setup
    const int ext  = CHUNK + 2 * halo;
    const long long own0 = (long long)blockIdx.x * CHUNK;
    const long long s    = own0 - halo;       // extended-region start (may be <0)

    for (int i = tid; i < T; i += 256) sbeta[i] = betas[i];

    for (int idx = tid; idx < ext; idx += 256) {
        long long g = s + idx;
        g = (g < 0) ? 0 : ((g >= N) ? (long long)(N - 1) : g);
        buf[0][idx] = I0[g];
    }

    const float dt      = t_steps[1] - t_steps[0];
    const float dx      = x_grid[1] - x_grid[0];
    const float inv_dx2 = 1.0f / (dx * dx);
    const float dp      = diff_param[0];
    const float spv     = (dp > 20.0f) ? dp : log1pf(expf(dp));
    const float diff    = fminf(fmaxf(spv, 0.0f), 1.0f);

    __syncthreads();

    int cur = 0;
    for (int t = 0; t < T; ++t) {
        const float beta = sbeta[t];
        const int nxt = cur ^ 1;
        float* __restrict__ rowp = history + (long long)t * N;

        for (int idx = tid; idx < ext; idx += 256) {
            const long long g = s + idx;
            const float c = buf[cur][idx];
            // replicated boundary at domain edges; buffer-edge clamp is
            // harmless (those cells are outside everyone's correct region)
            const int li = (idx > 0 && g > 0)           ? idx - 1 : idx;
            const int ri = (idx < ext - 1 && g < N - 1) ? idx + 1 : idx;
            const float l = buf[cur][li];
            const float r = buf[cur][ri];

            const float ixx  = (r - 2.0f * c + l) * inv_dx2;
            const float dIdt = diff * ixx + beta * (1.0f - c) * c - c;
            float v = c + dt * dIdt;
            v = fminf(fmaxf(v, 0.0f), 10.0f);

            buf[nxt][idx] = v;
            if (idx >= halo && idx < halo + CHUNK && g < N) {
                __builtin_nontemporal_store(v, &rowp[g]);
            }
        }
        cur = nxt;
        __syncthreads();
    }
}

// ---------------------------------------------------------------------------
// Inputs (setup_inputs order): t_steps[T], x_grid[N], initial_I[N], grid1[G],
// spline_w1[G*16], base_w1[16], grid2[G], spline_w2[16*G], base_w2[16],
// diff_param[1].  Output: history[T*N] f32.
// ---------------------------------------------------------------------------
extern "C" void kernel_launch(void* const* d_in, const int* in_sizes, int n_in,
                              void* d_out, int out_size, void* d_ws, size_t ws_size,
                              hipStream_t stream) {
    const float* t_steps   = (const float*)d_in[0];
    const float* x_grid    = (const float*)d_in[1];
    const float* initial_I = (const float*)d_in[2];
    const float* grid1     = (const float*)d_in[3];
    const float* sw1       = (const float*)d_in[4];
    const float* bw1       = (const float*)d_in[5];
    const float* grid2     = (const float*)d_in[6];
    const float* sw2       = (const float*)d_in[7];
    const float* bw2       = (const float*)d_in[8];
    const float* diff_p    = (const float*)d_in[9];

    const int T = in_sizes[0];   // 100
    const int N = in_sizes[1];   // 500000
    const int G = in_sizes[3];   // 20

    float* betas   = (float*)d_ws;
    float* history = (float*)d_out;

    kan_beta_kernel<<<1, 256, 0, stream>>>(t_steps, grid1, sw1, bw1,
                                           grid2, sw2, bw2, betas, T, G);

    const int nblocks = (N + CHUNK - 1) / CHUNK;   // 489
    pde_history_kernel<<<nblocks, 256, 0, stream>>>(initial_I, betas, t_steps,
                                                    x_grid, diff_p, history, N, T);
}